// NetGAT_86285892977177
// MI455X (gfx1250) — compile-verified
//
#include <hip/hip_runtime.h>
#include <math.h>

typedef __attribute__((ext_vector_type(2))) float v2f;
typedef __attribute__((ext_vector_type(8))) float v8f;

#define HEADS 4
#define HIDC  128
#define XHW   (HEADS * HIDC)   // 512
#define NEG_SLOPE 0.2f

// ---------- helpers ----------

__device__ __forceinline__ void atomicAddF(float* p, float v) {
  // native global_atomic_add_f32 on CDNA5
  unsafeAtomicAdd(p, v);
}

// monotone float max via int atomics (works for mixed signs, init = -inf)
__device__ __forceinline__ void atomicMaxF(float* addr, float v) {
  if (v >= 0.0f) atomicMax((int*)addr, __float_as_int(v));
  else           atomicMin((unsigned int*)addr, __float_as_uint(v));
}

__global__ void fill_f32(float* __restrict__ p, float v, int n) {
  int i = blockIdx.x * blockDim.x + threadIdx.x;
  if (i < n) p[i] = v;
}

// ---------- GEMM: C[M,512] = A[M,128] (row stride lda) x B[128,512] ----------
// One wave per 16x16 output tile; fp32 WMMA 16x16x4, K-loop of 32 steps.
__global__ __launch_bounds__(256) void gemm_xw_wmma(
    const float* __restrict__ A, int lda,
    const float* __restrict__ B,     // [K, Nn] row-major
    float* __restrict__ C,           // [M, Nn]
    int M, int K, int Nn)
{
  const int lane = threadIdx.x & 31;
  const int wid  = (blockIdx.x * blockDim.x + threadIdx.x) >> 5;
  const int tilesN = Nn >> 4;
  const int tiles  = (M >> 4) * tilesN;
  if (wid >= tiles) return;
  const int tm = wid / tilesN;
  const int tn = wid % tilesN;
  const int half = lane >> 4;       // 0: lanes 0-15, 1: lanes 16-31
  const int l16  = lane & 15;

  const float* arow = A + (size_t)(tm * 16 + l16) * lda;
  const float* bcol = B + (tn * 16 + l16);

  v8f acc = {};
  for (int k0 = 0; k0 < K; k0 += 4) {
    // A 16x4 f32 frag: lanes 0-15 -> K = k0+0,k0+1 ; lanes 16-31 -> K = k0+2,k0+3
    v2f af, bf;
    af.x = arow[k0 + 2 * half + 0];
    af.y = arow[k0 + 2 * half + 1];
    // B 4x16 f32 frag: row striped across lanes within a VGPR (mirror of A's K split)
    bf.x = bcol[(size_t)(k0 + 2 * half + 0) * Nn];
    bf.y = bcol[(size_t)(k0 + 2 * half + 1) * Nn];
    acc = __builtin_amdgcn_wmma_f32_16x16x4_f32(
        /*neg_a=*/false, af, /*neg_b=*/false, bf,
        /*c_mod=*/(short)0, acc, /*reuse_a=*/false, /*reuse_b=*/false);
  }
  // C/D: VGPR r, lanes 0-15 -> M = r, lanes 16-31 -> M = 8 + r ; N = l16
  float* crow = C + (size_t)(tm * 16 + half * 8) * Nn + tn * 16 + l16;
#pragma unroll
  for (int r = 0; r < 8; ++r) crow[(size_t)r * Nn] = acc[r];
}

// ---------- per-node attention dot products ----------
// one wave per (node, head): alpha_s[n,h] = <xh[n,h,:], a_src[h,:]>, same for dst
__global__ __launch_bounds__(256) void node_alpha(
    const float* __restrict__ xh,      // [N, 512]
    const float* __restrict__ a_src,   // [H, 128]
    const float* __restrict__ a_dst,   // [H, 128]
    float* __restrict__ alphaS, float* __restrict__ alphaD, int N)
{
  const int wid  = (blockIdx.x * blockDim.x + threadIdx.x) >> 5;
  const int lane = threadIdx.x & 31;
  if (wid >= N * HEADS) return;
  const int n = wid >> 2, h = wid & 3;
  const float* row = xh + (size_t)n * XHW + h * HIDC;
  const float* vs  = a_src + h * HIDC;
  const float* vd  = a_dst + h * HIDC;
  float ps = 0.f, pd = 0.f;
  for (int c = lane; c < HIDC; c += 32) {
    float v = row[c];
    ps += v * vs[c];
    pd += v * vd[c];
  }
  for (int off = 16; off; off >>= 1) {
    ps += __shfl_xor(ps, off, 32);
    pd += __shfl_xor(pd, off, 32);
  }
  if (lane == 0) { alphaS[wid] = ps; alphaD[wid] = pd; }
}

// ---------- edge logits + segment max ----------
__global__ __launch_bounds__(256) void edge_logits(
    const int* __restrict__ ei, int E, int Et,
    const float* __restrict__ alphaS, const float* __restrict__ alphaD,
    float* __restrict__ elog, float* __restrict__ m)
{
  int e = blockIdx.x * blockDim.x + threadIdx.x;
  if (e >= Et) return;
  int s, d;
  if (e < E) { s = ei[e]; d = ei[E + e]; } else { s = d = e - E; }
#pragma unroll
  for (int h = 0; h < HEADS; ++h) {
    float l = alphaS[s * HEADS + h] + alphaD[d * HEADS + h];
    l = (l >= 0.f) ? l : NEG_SLOPE * l;
    elog[(size_t)e * HEADS + h] = l;
    atomicMaxF(&m[d * HEADS + h], l);
  }
}

// ---------- exp(logit - m[dst]) + segment sum ----------
__global__ __launch_bounds__(256) void edge_exp(
    const int* __restrict__ ei, int E, int Et,
    const float* __restrict__ m,
    float* __restrict__ elog, float* __restrict__ denom)
{
  int e = blockIdx.x * blockDim.x + threadIdx.x;
  if (e >= Et) return;
  int d = (e < E) ? ei[E + e] : (e - E);
#pragma unroll
  for (int h = 0; h < HEADS; ++h) {
    float mm = m[d * HEADS + h];
    if (!__builtin_isfinite(mm)) mm = 0.0f;
    float v = expf(elog[(size_t)e * HEADS + h] - mm);
    elog[(size_t)e * HEADS + h] = v;
    atomicAddF(&denom[d * HEADS + h], v);
  }
}

// ---------- weighted aggregation: accum[dst] += alpha * xh[src] ----------
// one wave per edge; each lane handles 16 of the 512 channels
__global__ __launch_bounds__(256) void edge_aggregate(
    const int* __restrict__ ei, int E, int Et,
    const float* __restrict__ elog, const float* __restrict__ denom,
    const float* __restrict__ xh, float* __restrict__ accum)
{
  const int wid  = (blockIdx.x * blockDim.x + threadIdx.x) >> 5;
  const int lane = threadIdx.x & 31;
  if (wid >= Et) return;
  int s, d;
  if (wid < E) { s = ei[wid]; d = ei[E + wid]; } else { s = d = wid - E; }
  float a[HEADS];
#pragma unroll
  for (int h = 0; h < HEADS; ++h)
    a[h] = elog[(size_t)wid * HEADS + h] / denom[d * HEADS + h];
  const float* xs = xh + (size_t)s * XHW;
  float* acc      = accum + (size_t)d * XHW;
  for (int i = lane; i < XHW; i += 32)
    atomicAddF(&acc[i], a[i >> 7] * xs[i]);
}

// ---------- head mean + bias ----------
__global__ __launch_bounds__(256) void feats_finalize(
    const float* __restrict__ accum, const float* __restrict__ b,
    float* __restrict__ fout, int N)
{
  int t = blockIdx.x * blockDim.x + threadIdx.x;
  if (t >= N * HIDC) return;
  int n = t >> 7, c = t & 127;
  const float* p = accum + (size_t)n * XHW + c;
  fout[t] = 0.25f * (p[0] + p[HIDC] + p[2 * HIDC] + p[3 * HIDC]) + b[c];
}

// ---------- final pair scorer ----------
__global__ __launch_bounds__(256) void pair_predict(
    const float* __restrict__ feats, const float* __restrict__ x, int ldx,
    const int* __restrict__ idx, int P,
    const float* __restrict__ fc_w, const float* __restrict__ fc_b,
    float* __restrict__ out)
{
  const int wid  = (blockIdx.x * blockDim.x + threadIdx.x) >> 5;
  const int lane = threadIdx.x & 31;
  if (wid >= P) return;
  const int i1 = idx[wid], i2 = idx[P + wid];
  const float* f1 = feats + (size_t)i1 * HIDC;
  const float* f2 = feats + (size_t)i2 * HIDC;
  float p = 0.f;
  for (int c = lane; c < HIDC; c += 32) p += f1[c] * f2[c];
  for (int off = 16; off; off >>= 1) p += __shfl_xor(p, off, 32);
  if (lane == 0) {
    float dx = x[(size_t)i1 * ldx + 0] - x[(size_t)i2 * ldx + 0];
    float dy = x[(size_t)i1 * ldx + 1] - x[(size_t)i2 * ldx + 1];
    float pos = dx * dx + dy * dy;
    out[wid] = p * fc_w[0] + pos * fc_w[1] + fc_b[0];
  }
}

// ---------- host launcher ----------
extern "C" void kernel_launch(void* const* d_in, const int* in_sizes, int n_in,
                              void* d_out, int out_size, void* d_ws, size_t ws_size,
                              hipStream_t stream) {
  const float* x      = (const float*)d_in[0];   // [N, 130]
  const int*   ei     = (const int*)d_in[1];     // [2, E]
  const int*   idx    = (const int*)d_in[2];     // [2, P]
  const float* W1     = (const float*)d_in[3];
  const float* a_src1 = (const float*)d_in[4];
  const float* a_dst1 = (const float*)d_in[5];
  const float* b1     = (const float*)d_in[6];
  const float* W2     = (const float*)d_in[7];
  const float* a_src2 = (const float*)d_in[8];
  const float* a_dst2 = (const float*)d_in[9];
  const float* b2     = (const float*)d_in[10];
  const float* fc_w   = (const float*)d_in[11];
  const float* fc_b   = (const float*)d_in[12];

  const int XW = 2 + HIDC;              // 130
  const int N  = in_sizes[0] / XW;
  const int E  = in_sizes[1] / 2;
  const int P  = in_sizes[2] / 2;
  const int Et = E + N;

  // carve workspace
  char* w = (char*)d_ws;
  auto carve = [&](size_t bytes) -> void* {
    void* p = (void*)w;
    w += (bytes + 255) & ~(size_t)255;
    return p;
  };
  float* xh     = (float*)carve((size_t)N * XHW * sizeof(float));
  float* accum  = (float*)carve((size_t)N * XHW * sizeof(float));
  float* f1     = (float*)carve((size_t)N * HIDC * sizeof(float));
  float* f2     = (float*)carve((size_t)N * HIDC * sizeof(float));
  float* alphaS = (float*)carve((size_t)N * HEADS * sizeof(float));
  float* alphaD = (float*)carve((size_t)N * HEADS * sizeof(float));
  float* mbuf   = (float*)carve((size_t)N * HEADS * sizeof(float));
  float* denom  = (float*)carve((size_t)N * HEADS * sizeof(float));
  float* elog   = (float*)carve((size_t)Et * HEADS * sizeof(float));
  (void)ws_size; (void)n_in; (void)out_size;

  auto layer = [&](const float* A, int lda, const float* Wm,
                   const float* asv, const float* adv, const float* bv,
                   float* fout) {
    const int tiles = (N / 16) * (XHW / 16);
    gemm_xw_wmma<<<(tiles + 7) / 8, 256, 0, stream>>>(A, lda, Wm, xh, N, HIDC, XHW);
    fill_f32<<<((N * HEADS) + 255) / 256, 256, 0, stream>>>(mbuf, -INFINITY, N * HEADS);
    fill_f32<<<((N * HEADS) + 255) / 256, 256, 0, stream>>>(denom, 0.0f, N * HEADS);
    fill_f32<<<((N * XHW) + 255) / 256, 256, 0, stream>>>(accum, 0.0f, N * XHW);
    node_alpha<<<((N * HEADS) + 7) / 8, 256, 0, stream>>>(xh, asv, adv, alphaS, alphaD, N);
    edge_logits<<<(Et + 255) / 256, 256, 0, stream>>>(ei, E, Et, alphaS, alphaD, elog, mbuf);
    edge_exp<<<(Et + 255) / 256, 256, 0, stream>>>(ei, E, Et, mbuf, elog, denom);
    edge_aggregate<<<(Et + 7) / 8, 256, 0, stream>>>(ei, E, Et, elog, denom, xh, accum);
    feats_finalize<<<((N * HIDC) + 255) / 256, 256, 0, stream>>>(accum, bv, fout, N);
  };

  // layer 1: feats = x[:, 2:130] (row stride 130)
  layer(x + 2, XW, W1, a_src1, a_dst1, b1, f1);
  // layer 2
  layer(f1, HIDC, W2, a_src2, a_dst2, b2, f2);
  // final pairwise scorer
  pair_predict<<<(P + 7) / 8, 256, 0, stream>>>(f2, x, XW, idx, P, fc_w, fc_b, (float*)d_out);
}